// SingleViewNet_7765300871556
// MI455X (gfx1250) — compile-verified
//
#include <hip/hip_runtime.h>
#include <hip/hip_bf16.h>

// ---------------- problem constants (match reference) ----------------
#define BB   2
#define PP   100000
#define CIN  23
#define CC   64
#define HH   512
#define WW   512
#define SS   (HH * WW)          // 262144 voxels per batch
#define NROWS (BB * SS)         // 524288 total voxel rows
#define EPS_BN1 1e-5f
#define EPS_BN2 1e-3f

typedef __attribute__((ext_vector_type(2))) float v2f;
typedef __attribute__((ext_vector_type(8))) float v8f;

// =====================================================================
// Kernel 1: PointNet (Linear 23->64, BN1, ReLU) + scatter segment-max.
// Voxel buffer is pre-zeroed; all contributions are >= 0 after ReLU, so
// uint atomicMax on float bits == float segment max with empty->0.
// =====================================================================
__global__ __launch_bounds__(256)
void pointnet_scatter_kernel(const float* __restrict__ feat,    // [B*P, 23]
                             const int*   __restrict__ indices, // [B*P]
                             const int*   __restrict__ padd,    // [B*P]
                             const float* __restrict__ w_pn,    // [23, 64]
                             const float* __restrict__ g1,
                             const float* __restrict__ b1,
                             const float* __restrict__ m1,
                             const float* __restrict__ v1,
                             float*       __restrict__ voxels)  // [NROWS, 64]
{
    __shared__ float Wl[CIN * CC];   // 23*64 floats, addr = i*64+o -> conflict-free
    __shared__ float s_sc[CC], s_bi[CC];

    for (int i = threadIdx.x; i < CIN * CC; i += 256) Wl[i] = w_pn[i];
    if (threadIdx.x < CC) {
        const int o  = threadIdx.x;
        const float inv = rsqrtf(v1[o] + EPS_BN1);
        const float s   = inv * g1[o];
        s_sc[o] = s;
        s_bi[o] = b1[o] - m1[o] * s;
    }
    __syncthreads();

    const int p = blockIdx.x * 256 + threadIdx.x;
    if (p >= BB * PP) return;
    if (padd[p] != 0) return;                 // invalid point: contributes only 0

    float f[CIN];
#pragma unroll
    for (int i = 0; i < CIN; ++i) f[i] = feat[p * CIN + i];

    const int b   = p / PP;
    const int row = b * SS + indices[p];
    unsigned int* dst = reinterpret_cast<unsigned int*>(voxels) + row * CC;

#pragma unroll 4
    for (int o = 0; o < CC; ++o) {
        float acc = 0.0f;
#pragma unroll
        for (int i = 0; i < CIN; ++i) acc = fmaf(f[i], Wl[i * CC + o], acc);
        float y = fmaxf(fmaf(acc, s_sc[o], s_bi[o]), 0.0f);
        if (y > 0.0f) atomicMax(dst + o, __float_as_uint(y));  // uint order == float order for y>=0
    }
}

// =====================================================================
// Kernel 2: dense 1x1 conv (GEMM [NROWS x 64] * W^T[64 x 64]) + BN2 +
// ReLU, IN-PLACE on the voxel buffer, using V_WMMA_F32_16X16X4_F32.
// One wave -> 16 rows x 64 channels; 8 waves/block -> 128 rows/block.
// =====================================================================
__global__ __launch_bounds__(256)
void conv_bn_relu_kernel(float* __restrict__ X,              // [NROWS, 64] in/out
                         const float* __restrict__ conv_w,   // [64(out), 64(in)]
                         const float* __restrict__ g2,
                         const float* __restrict__ b2,
                         const float* __restrict__ m2,
                         const float* __restrict__ v2)
{
    __shared__ float Wl[CC * 65];      // padded stride 65: bank-conflict-free B reads
    __shared__ float s_sc[CC], s_bi[CC];

    for (int i = threadIdx.x; i < CC * CC; i += 256) {
        const int o = i >> 6, c = i & 63;
        Wl[o * 65 + c] = conv_w[i];
    }
    if (threadIdx.x < CC) {
        const int o  = threadIdx.x;
        const float inv = rsqrtf(v2[o] + EPS_BN2);
        const float s   = inv * g2[o];
        s_sc[o] = s;
        s_bi[o] = b2[o] - m2[o] * s;
    }
    __syncthreads();

    const int lane = threadIdx.x & 31;
    const int wave = threadIdx.x >> 5;
    const int rowBase = (blockIdx.x * 8 + wave) * 16;   // <= 524272, fits int

    const int mlo = lane & 15;          // A: M index  /  B,C,D: N index (mod 16)
    const int kb  = (lane >> 4) << 1;   // half-wave K pair offset {0,2}

    // ---- A fragments: 16 K-chunks of 16x4 fp32 (lane m = mlo, K = kc*4+kb{,+1})
    const float* Xrow = X + (rowBase + mlo) * CC;
    v2f a[16];
#pragma unroll
    for (int kc = 0; kc < 16; ++kc) {
        const int k = (kc << 2) + kb;
        a[kc].x = Xrow[k];
        a[kc].y = Xrow[k + 1];
    }

    // ---- 4 N-tiles x 16 K-steps of v_wmma_f32_16x16x4_f32
    v8f acc[4] = {v8f{}, v8f{}, v8f{}, v8f{}};
#pragma unroll
    for (int nt = 0; nt < 4; ++nt) {
        const int n = (nt << 4) + mlo;                  // output channel for this lane col
#pragma unroll
        for (int kc = 0; kc < 16; ++kc) {
            const int k = (kc << 2) + kb;
            v2f bfrag;
            bfrag.x = Wl[n * 65 + k];                   // B[k,n]   = W[n,k]
            bfrag.y = Wl[n * 65 + k + 1];               // B[k+1,n] = W[n,k+1]
            acc[nt] = __builtin_amdgcn_wmma_f32_16x16x4_f32(
                false, a[kc], false, bfrag, (short)0, acc[nt], false, false);
        }
    }

    // ---- epilogue: BN2 + ReLU, in-place store (this wave owns these 16 rows)
#pragma unroll
    for (int nt = 0; nt < 4; ++nt) {
        const int ch = (nt << 4) + mlo;
        const float sc = s_sc[ch];
        const float bi = s_bi[ch];
#pragma unroll
        for (int r = 0; r < 8; ++r) {
            const int m = r + ((lane >> 4) << 3);       // D layout: VGPR r -> M=r / r+8
            float y = fmaxf(fmaf(acc[nt][r], sc, bi), 0.0f);
            X[(rowBase + m) * CC + ch] = y;
        }
    }
}

// =====================================================================
// Kernel 3: bilinear gather. im[b, x, y, c] = V[b, y, x, c].
// One thread per (point, channel): 64 threads/point -> 256B coalesced
// corner reads (voxel buffer is L2-resident: 134MB < 192MB L2).
// =====================================================================
__global__ __launch_bounds__(256)
void bilinear_gather_kernel(const float* __restrict__ V,     // [NROWS, 64] conv output
                            const float* __restrict__ vxyz,  // [B*P, 3]
                            float*       __restrict__ out)   // [B*P, 64]
{
    const int t = blockIdx.x * 256 + threadIdx.x;
    const int c = t & 63;
    const int p = t >> 6;
    if (p >= BB * PP) return;

    const int b  = p / PP;
    const float xq = vxyz[p * 3 + 0];
    const float yq = vxyz[p * 3 + 1];

    int x0 = (int)floorf(xq);
    x0 = min(max(x0, 0), WW - 1);
    const int x1 = min(x0 + 1, WW - 1);
    int y0 = (int)floorf(yq);
    y0 = min(max(y0, 0), HH - 1);
    const int y1 = min(y0 + 1, HH - 1);

    const float x0f = (float)x0, x1f = (float)x1;
    const float y0f = (float)y0, y1f = (float)y1;
    const float wa = (x1f - xq) * (y1f - yq);
    const float wb = (x1f - xq) * (yq - y0f);
    const float wc = (xq - x0f) * (y1f - yq);
    const float wd = (xq - x0f) * (yq - y0f);

    const int base = b * SS;
    const float Ia = V[(base + y0 * WW + x0) * CC + c];
    const float Ib = V[(base + y1 * WW + x0) * CC + c];
    const float Ic = V[(base + y0 * WW + x1) * CC + c];
    const float Id = V[(base + y1 * WW + x1) * CC + c];

    out[p * CC + c] = wa * Ia + wb * Ib + wc * Ic + wd * Id;
}

// =====================================================================
// launch
// =====================================================================
extern "C" void kernel_launch(void* const* d_in, const int* in_sizes, int n_in,
                              void* d_out, int out_size, void* d_ws, size_t ws_size,
                              hipStream_t stream)
{
    // setup_inputs() order:
    // 0 points_xyz  1 points_feature  2 points_mask  3 indices  4 paddings
    // 5 voxel_xyz   6 w_pn  7 bn1_gamma 8 bn1_beta 9 bn1_mean 10 bn1_var
    // 11 conv_w     12 bn2_gamma 13 bn2_beta 14 bn2_mean 15 bn2_var
    const float* feat    = (const float*)d_in[1];
    const int*   indices = (const int*)  d_in[3];
    const int*   padd    = (const int*)  d_in[4];
    const float* vxyz    = (const float*)d_in[5];
    const float* w_pn    = (const float*)d_in[6];
    const float* g1      = (const float*)d_in[7];
    const float* b1      = (const float*)d_in[8];
    const float* m1      = (const float*)d_in[9];
    const float* v1      = (const float*)d_in[10];
    const float* conv_w  = (const float*)d_in[11];
    const float* g2      = (const float*)d_in[12];
    const float* b2      = (const float*)d_in[13];
    const float* m2      = (const float*)d_in[14];
    const float* v2      = (const float*)d_in[15];

    float* vox = (float*)d_ws;                       // [NROWS, 64] = 134 MB
    const size_t voxBytes = (size_t)NROWS * CC * sizeof(float);

    // stage 0: zero voxel grid (segment-max identity, since all updates >= 0)
    hipMemsetAsync(vox, 0, voxBytes, stream);

    // stage 1: pointnet + scatter max
    pointnet_scatter_kernel<<<(BB * PP + 255) / 256, 256, 0, stream>>>(
        feat, indices, padd, w_pn, g1, b1, m1, v1, vox);

    // stage 2: in-place 1x1 conv + BN2 + ReLU via fp32 WMMA (128 rows/block)
    conv_bn_relu_kernel<<<NROWS / 128, 256, 0, stream>>>(
        vox, conv_w, g2, b2, m2, v2);

    // stage 3: bilinear gather -> output [B, P, 64]
    bilinear_gather_kernel<<<(BB * PP * CC + 255) / 256, 256, 0, stream>>>(
        vox, vxyz, (float*)d_out);
}